// Conduits_8899172237900
// MI455X (gfx1250) — compile-verified
//
#include <hip/hip_runtime.h>

// ---------------------------------------------------------------------------
// Glacier conduit network step, MI455X (gfx1250, wave32).
// HBM-bound gather/stream workload (~0.45 GB/step, <0.1 FLOP/B).
//   Pass 1 (links): compute {discharge, dissipation} once per link, packed as
//     float2 in d_ws. Head/tail index streams are double-buffered one
//     grid-stride ahead via global_load_async_to_lds_b32 (ASYNCcnt), since
//     they feed the dependent gathers. NT hints on all streamed-once data.
//   Pass 2 (nodes): 6x random float2 (b64) gathers of the L2-resident
//     intermediate; index rows loaded as 3x b64 NT; NT final stores.
// The 32MB intermediate stays hot in the 192MB L2 because everything that is
// read exactly once carries th:NT and outputs are NT stores.
// ---------------------------------------------------------------------------

#define G_ACC   9.81f
#define RHO_I   917.0f
#define RHO_W   1000.0f
#define NU_W    1.787e-6f
#define OMEGA   1e-3f
#define L_HEAT  334000.0f
#define A_ICE   6e-24f
#define DT_STEP 0.01f

typedef int   v2i __attribute__((ext_vector_type(2)));
typedef float v2f __attribute__((ext_vector_type(2)));

// Lane-private LDS slot address as a 32-bit LDS byte offset (low 32 bits of
// the generic pointer to a __shared__ object are the wave-relative LDS addr).
__device__ __forceinline__ unsigned lds_off(const void* p) {
    return (unsigned)(uintptr_t)p;
}

// ---- Pass 1: per-link transmissivity / discharge / dissipation -------------
__global__ __launch_bounds__(256) void conduits_link_kernel(
    const float* __restrict__ conduit_size,     // [N]
    const float* __restrict__ hydraulic_head,   // [N]
    const float* __restrict__ reynolds,         // [L] streamed once
    const float* __restrict__ length_of_link,   // [L] streamed once
    const int*   __restrict__ node_head,        // [L] streamed once (async->LDS)
    const int*   __restrict__ node_tail,        // [L] streamed once (async->LDS)
    v2f*         __restrict__ q_diss,           // [L] intermediate (keep in L2)
    int L)
{
    __shared__ int s_head[2][256];
    __shared__ int s_tail[2][256];

    const int stride = (int)(gridDim.x * blockDim.x);
    const int tid    = (int)threadIdx.x;
    const int i0     = (int)(blockIdx.x * blockDim.x) + tid;

    const unsigned lh[2] = { lds_off(&s_head[0][tid]), lds_off(&s_head[1][tid]) };
    const unsigned lt[2] = { lds_off(&s_tail[0][tid]), lds_off(&s_tail[1][tid]) };

    // Prologue: stage tile 0 into buffer 0 (ASYNCcnt += 2 for this wave).
    if (i0 < L) {
        const unsigned voff = (unsigned)i0 * 4u;
        asm volatile("global_load_async_to_lds_b32 %0, %1, %2 th:TH_LOAD_NT"
                     :: "v"(lh[0]), "v"(voff), "s"(node_head) : "memory");
        asm volatile("global_load_async_to_lds_b32 %0, %1, %2 th:TH_LOAD_NT"
                     :: "v"(lt[0]), "v"(voff), "s"(node_tail) : "memory");
    }

    int par = 0;
    for (int i = i0; i < L; i += stride) {
        // Stage the NEXT tile (clamped so every iteration issues exactly two
        // async loads -> per-wave ASYNCcnt bookkeeping stays uniform).
        {
            const int nxt = i + stride;
            const int nc  = (nxt < L) ? nxt : i;
            const unsigned voffn = (unsigned)nc * 4u;
            asm volatile("global_load_async_to_lds_b32 %0, %1, %2 th:TH_LOAD_NT"
                         :: "v"(lh[par ^ 1]), "v"(voffn), "s"(node_head) : "memory");
            asm volatile("global_load_async_to_lds_b32 %0, %1, %2 th:TH_LOAD_NT"
                         :: "v"(lt[par ^ 1]), "v"(voffn), "s"(node_tail) : "memory");
        }
        // Async loads complete in order: <=2 outstanding means the current
        // tile's two loads have landed in LDS; the next tile stays in flight.
        asm volatile("s_wait_asynccnt 0x2" ::: "memory");

        const int   h  = s_head[par][tid];
        const int   t  = s_tail[par][tid];
        const float re = __builtin_nontemporal_load(&reynolds[i]);
        const float ll = __builtin_nontemporal_load(&length_of_link[i]);

        // Gathers: 8MB arrays, reused ~2x across links + again in pass 2;
        // regular (RT) policy so they cache.
        const float s_link = 0.5f * (conduit_size[h] + conduit_size[t]);
        const float grad_h = (hydraulic_head[h] - hydraulic_head[t]) / ll;

        const float transmissivity =
            (s_link * s_link * s_link) * G_ACC / (12.0f * NU_W * (1.0f + OMEGA * re));
        const float q = transmissivity * grad_h;

        // Packed pair -> pass 2 needs ONE b64 gather per link reference.
        // Default (RT) store policy: we WANT this resident in the 192MB L2.
        v2f out; out.x = q; out.y = fabsf(q * grad_h);
        q_diss[i] = out;

        par ^= 1;
    }
}

// ---- Pass 2: per-node divergence, melt, creep closure, RK4 -----------------
__global__ __launch_bounds__(256) void conduits_node_kernel(
    const float* __restrict__ conduit_size,      // [N]
    const float* __restrict__ hydraulic_head,    // [N]
    const float* __restrict__ ice_thickness,     // [N] streamed once
    const float* __restrict__ bedrock_elev,      // [N] streamed once
    const float* __restrict__ geothermal,        // [N] streamed once
    const float* __restrict__ area_at_node,      // [N] streamed once
    const int*   __restrict__ links_at_node,     // [N*6] streamed once
    const int*   __restrict__ link_dirs,         // [N*6] streamed once
    const v2f*   __restrict__ q_diss,            // [L] random gather, L2-hot
    float*       __restrict__ out,               // [2*N]
    int N)
{
    const int stride = (int)(gridDim.x * blockDim.x);
    for (int i = (int)(blockIdx.x * blockDim.x + threadIdx.x); i < N; i += stride) {
        const int base = i * 6;
        if (i + stride < N) {
            __builtin_prefetch(&links_at_node[base + stride * 6], 0, 0);
            __builtin_prefetch(&link_dirs[base + stride * 6], 0, 0);
        }

        // Row offset i*24B is always 8-byte aligned -> 3x b64 NT per array.
        const v2i* lan2 = (const v2i*)(links_at_node + base);
        const v2i* dir2 = (const v2i*)(link_dirs + base);
        int lk[6], dr[6];
#pragma unroll
        for (int k = 0; k < 3; ++k) {
            const v2i a = __builtin_nontemporal_load(&lan2[k]);
            const v2i d = __builtin_nontemporal_load(&dir2[k]);
            lk[2 * k] = a.x; lk[2 * k + 1] = a.y;
            dr[2 * k] = d.x; dr[2 * k + 1] = d.y;
        }

        float flux_sum = 0.0f;
        float diss_sum = 0.0f;
#pragma unroll
        for (int k = 0; k < 6; ++k) {
            const v2f v = q_diss[lk[k]];       // single b64 gather, hits L2
            flux_sum += (float)dr[k] * v.x;
            diss_sum += v.y;                   // mean includes dir==0 links
        }

        const float area  = __builtin_nontemporal_load(&area_at_node[i]);
        const float geo   = __builtin_nontemporal_load(&geothermal[i]);
        const float thick = __builtin_nontemporal_load(&ice_thickness[i]);
        const float bed   = __builtin_nontemporal_load(&bedrock_elev[i]);
        const float head  = hydraulic_head[i];  // reused from pass 1 -> L2 hit
        const float S     = conduit_size[i];    // reused from pass 1 -> L2 hit

        const float flux_div = flux_sum / area;

        const float diss_node = RHO_W * G_ACC * (diss_sum * (1.0f / 6.0f));
        const float melt_rate = (geo + diss_node) / L_HEAT;
        const float melt_term = melt_rate * (1.0f / RHO_W - 1.0f / RHO_I);

        const float overburden = RHO_I * G_ACC * thick;
        const float water_p    = RHO_W * G_ACC * (head - bed);
        const float n_eff      = overburden - water_p;
        const float closure    = A_ICE * (n_eff * n_eff) * n_eff;  // signed cube

        // RK4 on dS/dt = melt_term - closure * S
        const float k1 = melt_term - closure * S;
        const float k2 = melt_term - closure * (S + k1 * (DT_STEP * 0.5f));
        const float k3 = melt_term - closure * (S + k2 * (DT_STEP * 0.5f));
        const float k4 = melt_term - closure * (S + k3 * DT_STEP);
        const float S_new =
            S + DT_STEP * (k1 + 2.0f * k2 + 2.0f * k3 + k4) * (1.0f / 6.0f);

        // Outputs are never re-read: non-temporal stores, don't pollute L2.
        __builtin_nontemporal_store(S_new,    &out[i]);
        __builtin_nontemporal_store(flux_div, &out[N + i]);
    }
}

extern "C" void kernel_launch(void* const* d_in, const int* in_sizes, int n_in,
                              void* d_out, int out_size, void* d_ws, size_t ws_size,
                              hipStream_t stream) {
    const float* conduit_size   = (const float*)d_in[0];
    const float* hydraulic_head = (const float*)d_in[1];
    const float* reynolds       = (const float*)d_in[2];
    const float* ice_thickness  = (const float*)d_in[3];
    const float* bedrock_elev   = (const float*)d_in[4];
    const float* geothermal     = (const float*)d_in[5];
    const float* length_of_link = (const float*)d_in[6];
    const float* area_at_node   = (const float*)d_in[7];
    const int*   node_head      = (const int*)d_in[8];
    const int*   node_tail      = (const int*)d_in[9];
    const int*   links_at_node  = (const int*)d_in[10];
    const int*   link_dirs      = (const int*)d_in[11];

    const int N = in_sizes[0];   // nodes
    const int L = in_sizes[2];   // links

    v2f*   q_diss = (v2f*)d_ws;  // 2*L floats = 32 MB, fits in ws and in L2
    float* out    = (float*)d_out;

    const int tb = 256;  // 8 wave32 per workgroup
    int b1 = (L + tb - 1) / tb; if (b1 > 4096) b1 = 4096;  // ~4 iters/thread
    int b2 = (N + tb - 1) / tb; if (b2 > 4096) b2 = 4096;  // ~2 iters/thread

    conduits_link_kernel<<<b1, tb, 0, stream>>>(
        conduit_size, hydraulic_head, reynolds, length_of_link,
        node_head, node_tail, q_diss, L);

    conduits_node_kernel<<<b2, tb, 0, stream>>>(
        conduit_size, hydraulic_head, ice_thickness, bedrock_elev,
        geothermal, area_at_node, links_at_node, link_dirs,
        q_diss, out, N);
}